// MemoryTapAssembly_58540404245066
// MI455X (gfx1250) — compile-verified
//
#include <hip/hip_runtime.h>
#include <math.h>

// Problem constants (from the reference): iq_sequence (64, 16384, 2) f32.
#define B_DIM     64
#define S_DIM     16384
#define MDEPTH    5
#define T_DIM     (S_DIM - MDEPTH)       // 16379
#define OUT_CH    18
#define BLOCK_T   256
#define NTHREADS  256
#define NLOAD_MAX (BLOCK_T + MDEPTH)     // 261 input positions per tile (halo)

typedef float v2f __attribute__((ext_vector_type(2)));
typedef int   v2i __attribute__((vector_size(8)));   // matches builtin param type

// CDNA5 async global<->LDS paths (ASYNCcnt), guarded so compile never breaks.
#if defined(__has_builtin)
#if __has_builtin(__builtin_amdgcn_global_load_async_to_lds_b64) && \
    __has_builtin(__builtin_amdgcn_s_wait_asynccnt)
#define USE_ASYNC_LD 1
#endif
#if __has_builtin(__builtin_amdgcn_global_store_async_from_lds_b64)
#define USE_ASYNC_ST 1
#endif
#endif
#ifndef USE_ASYNC_LD
#define USE_ASYNC_LD 0
#endif
#ifndef USE_ASYNC_ST
#define USE_ASYNC_ST 0
#endif

__global__ __launch_bounds__(NTHREADS) void memtap_kernel(
    const v2f* __restrict__ iq, float* __restrict__ out)
{
  __shared__ v2f s_iq[NLOAD_MAX];                 // staged input window (I,Q)
  __shared__ v2f s_out2[BLOCK_T * OUT_CH / 2];    // staged output tile (as float2)

  const int tid   = threadIdx.x;
  const int t0    = blockIdx.x * BLOCK_T;         // first output step of this tile
  const int b     = blockIdx.y;
  const int cnt   = min(BLOCK_T, T_DIM - t0);     // valid output rows this tile
  const int nload = min(NLOAD_MAX, S_DIM - t0);   // input positions to stage
  const size_t gbase = (size_t)b * S_DIM + (size_t)t0;

  // ---- Phase A: stage input window into LDS (async global->LDS) ----------
#if USE_ASYNC_LD
  if (tid < nload) {
    __builtin_amdgcn_global_load_async_to_lds_b64(
        (v2i*)(iq + gbase + tid), (v2i*)&s_iq[tid], 0, 0);
  }
  if (tid + NTHREADS < nload) {
    __builtin_amdgcn_global_load_async_to_lds_b64(
        (v2i*)(iq + gbase + tid + NTHREADS), (v2i*)&s_iq[tid + NTHREADS], 0, 0);
  }
  __builtin_amdgcn_s_wait_asynccnt(0);
  __syncthreads();
#else
  for (int i = tid; i < nload; i += NTHREADS)
    s_iq[i] = __builtin_nontemporal_load(&iq[gbase + i]);
  __syncthreads();
#endif

  // ---- Phase B: per-row assembly into LDS staging tile --------------------
  // Output row t (n = t+5):
  //   [0:2)  = I,Q at n
  //   [2:8)  = env(n), env(n-1), ..., env(n-5)
  //   [8:18) = I,Q at n-1, n-2, ..., n-5
  float* so = (float*)s_out2;
  if (tid < cnt) {
    const v2f w0 = s_iq[tid + 5];   // n
    const v2f w1 = s_iq[tid + 4];   // n-1
    const v2f w2 = s_iq[tid + 3];   // n-2
    const v2f w3 = s_iq[tid + 2];   // n-3
    const v2f w4 = s_iq[tid + 1];   // n-4
    const v2f w5 = s_iq[tid + 0];   // n-5
    float* o = so + tid * OUT_CH;
    o[0]  = w0.x;  o[1]  = w0.y;
    o[2]  = sqrtf(w0.x * w0.x + w0.y * w0.y);
    o[3]  = sqrtf(w1.x * w1.x + w1.y * w1.y);
    o[4]  = sqrtf(w2.x * w2.x + w2.y * w2.y);
    o[5]  = sqrtf(w3.x * w3.x + w3.y * w3.y);
    o[6]  = sqrtf(w4.x * w4.x + w4.y * w4.y);
    o[7]  = sqrtf(w5.x * w5.x + w5.y * w5.y);
    o[8]  = w1.x;  o[9]  = w1.y;
    o[10] = w2.x;  o[11] = w2.y;
    o[12] = w3.x;  o[13] = w3.y;
    o[14] = w4.x;  o[15] = w4.y;
    o[16] = w5.x;  o[17] = w5.y;
  }
  __syncthreads();   // all LDS writes visible before LDS->global streaming

  // ---- Phase C: coalesced streaming store of the contiguous tile ----------
  // Row base (b*T + t0)*18 is always even -> 8-byte aligned b64 stores.
  v2f* og = (v2f*)(out + ((size_t)b * T_DIM + (size_t)t0) * OUT_CH);
  const int tot2 = cnt * (OUT_CH / 2);            // cnt*9 float2 elements
#if USE_ASYNC_ST
  for (int j = tid; j < tot2; j += NTHREADS)
    __builtin_amdgcn_global_store_async_from_lds_b64(
        (v2i*)(og + j), (v2i*)&s_out2[j], 0, 0);
  // ASYNCcnt drains at the implicit S_WAIT_IDLE before S_ENDPGM.
#else
  for (int j = tid; j < tot2; j += NTHREADS)
    __builtin_nontemporal_store(s_out2[j], og + j);
#endif
}

extern "C" void kernel_launch(void* const* d_in, const int* in_sizes, int n_in,
                              void* d_out, int out_size, void* d_ws, size_t ws_size,
                              hipStream_t stream) {
  (void)in_sizes; (void)n_in; (void)out_size; (void)d_ws; (void)ws_size;
  const v2f* iq = (const v2f*)d_in[0];
  float* out = (float*)d_out;
  dim3 grid((T_DIM + BLOCK_T - 1) / BLOCK_T, B_DIM);  // (64, 64)
  memtap_kernel<<<grid, dim3(NTHREADS), 0, stream>>>(iq, out);
}